// SAGE_81793357185247
// MI455X (gfx1250) — compile-verified
//
#include <hip/hip_runtime.h>

typedef __attribute__((ext_vector_type(2))) float v2f;
typedef __attribute__((ext_vector_type(8))) float v8f;

static constexpr int NN = 50000;   // nodes
static constexpr int NE = 600000;  // edges
static constexpr int D0 = 128;     // in dim
static constexpr int D1 = 256;     // hidden
static constexpr int D2 = 64;      // out dim

// ---------------------------------------------------------------------------
// Zero a contiguous float range (agg1 | cnt | agg2 live contiguously in ws).
// ---------------------------------------------------------------------------
__global__ void sage_zero(float* __restrict__ p, long n) {
  long i = (long)blockIdx.x * blockDim.x + threadIdx.x;
  if (i < n) p[i] = 0.0f;
}

// ---------------------------------------------------------------------------
// Edge scatter: agg[dst] += feat[src] (float4 chunks), optional degree count.
// L2-resident: feat (<=51MB) fits the 192MB global L2, so the random gathers
// and atomic RMWs stay on-chip.
// ---------------------------------------------------------------------------
__global__ void sage_scatter(const float* __restrict__ feat,
                             const int* __restrict__ edges,  // [2, NE] flat
                             float* __restrict__ agg,
                             float* __restrict__ cnt,
                             int dim, int do_cnt) {
  const int chunks = dim >> 2;                  // float4 chunks per row
  long i = (long)blockIdx.x * blockDim.x + threadIdx.x;
  long total = (long)NE * chunks;
  if (i >= total) return;
  int e = (int)(i / chunks);
  int f = (int)(i % chunks) << 2;
  int src = edges[e];
  int dst = edges[NE + e];
  const float4 v = *(const float4*)(feat + (long)src * dim + f);
  float* a = agg + (long)dst * dim + f;
  atomicAdd(a + 0, v.x);
  atomicAdd(a + 1, v.y);
  atomicAdd(a + 2, v.z);
  atomicAdd(a + 3, v.w);
  if (do_cnt && f == 0) atomicAdd(cnt + dst, 1.0f);
}

// ---------------------------------------------------------------------------
// Fused SAGEConv GEMM using fp32 WMMA (V_WMMA_F32_16X16X4_F32):
//   out[16 rows x 64 cols per wave] =
//     (agg/deg) @ Wl + self @ Wr + bias   [+ relu]
// One wave (32 threads) per block; 4 f32 accumulators share the A fragment.
//
// A fragment (f32 16x4, ISA layout): lanes 0-15 hold K=k,k+1 for M=lane;
// lanes 16-31 hold K=k+2,k+3 for M=lane-16.  B mirrors with K rows striped
// across lanes.  C/D: VGPR i = row i (lanes 0-15) / row i+8 (lanes 16-31).
// ---------------------------------------------------------------------------
__global__ __launch_bounds__(32)
void sage_wmma_gemm(const float* __restrict__ Aagg,
                    const float* __restrict__ cnt,
                    const float* __restrict__ Aself,
                    const float* __restrict__ Wl,
                    const float* __restrict__ Wr,
                    const float* __restrict__ bias,
                    float* __restrict__ out,
                    int K, int Nout, int do_relu) {
  const int lane = threadIdx.x;
  const int half = lane >> 4;              // 0: K=k,k+1   1: K=k+2,k+3
  const int l15  = lane & 15;
  const int rowA = blockIdx.x * 16 + l15;  // A row held by this lane
  const int colBase = blockIdx.y * 64;

  const float scale = 1.0f / fmaxf(cnt[rowA], 1.0f);  // mean aggregation

  v8f cacc[4];
  #pragma unroll
  for (int t = 0; t < 4; ++t) cacc[t] = v8f{};

  for (int k = 0; k < K; k += 4) {
    const int k0 = k + half * 2;

    v2f a;  // aggregated-neighbor path, scaled to mean
    a.x = Aagg[(long)rowA * K + k0]     * scale;
    a.y = Aagg[(long)rowA * K + k0 + 1] * scale;
    v2f s;  // self (root-weight) path
    s.x = Aself[(long)rowA * K + k0];
    s.y = Aself[(long)rowA * K + k0 + 1];

    #pragma unroll
    for (int t = 0; t < 4; ++t) {
      const int col = colBase + t * 16 + l15;
      v2f bl;
      bl.x = Wl[(long)k0 * Nout + col];
      bl.y = Wl[(long)(k0 + 1) * Nout + col];
      cacc[t] = __builtin_amdgcn_wmma_f32_16x16x4_f32(
          false, a, false, bl, (short)0, cacc[t], false, false);
      v2f br;
      br.x = Wr[(long)k0 * Nout + col];
      br.y = Wr[(long)(k0 + 1) * Nout + col];
      cacc[t] = __builtin_amdgcn_wmma_f32_16x16x4_f32(
          false, s, false, br, (short)0, cacc[t], false, false);
    }
  }

  #pragma unroll
  for (int t = 0; t < 4; ++t) {
    const int col = colBase + t * 16 + l15;
    const float bcol = bias[col];
    #pragma unroll
    for (int i = 0; i < 8; ++i) {
      const int row = blockIdx.x * 16 + i + half * 8;
      float v = cacc[t][i] + bcol;
      if (do_relu) v = fmaxf(v, 0.0f);
      out[(long)row * Nout + col] = v;
    }
  }
}

// ---------------------------------------------------------------------------
extern "C" void kernel_launch(void* const* d_in, const int* in_sizes, int n_in,
                              void* d_out, int out_size, void* d_ws, size_t ws_size,
                              hipStream_t stream) {
  const float* x   = (const float*)d_in[0];
  const int*   ei  = (const int*)d_in[1];
  const float* W1l = (const float*)d_in[2];
  const float* W1r = (const float*)d_in[3];
  const float* b1  = (const float*)d_in[4];
  const float* W2l = (const float*)d_in[5];
  const float* W2r = (const float*)d_in[6];
  const float* b2  = (const float*)d_in[7];
  float* out = (float*)d_out;

  // Scratch layout (floats): agg1 | cnt | agg2 | h   (first 3 zero-initialized)
  float* ws   = (float*)d_ws;
  float* agg1 = ws;                          // NN*D0
  float* cnt  = agg1 + (size_t)NN * D0;      // NN
  float* agg2 = cnt + NN;                    // NN*D1
  float* h    = agg2 + (size_t)NN * D1;      // NN*D1

  const long zlen = (long)NN * D0 + NN + (long)NN * D1;
  sage_zero<<<(int)((zlen + 255) / 256), 256, 0, stream>>>(ws, zlen);

  // Layer 1 scatter: agg1 += x[src], cnt = degrees
  {
    long total = (long)NE * (D0 / 4);
    sage_scatter<<<(int)((total + 255) / 256), 256, 0, stream>>>(
        x, ei, agg1, cnt, D0, 1);
  }

  // Layer 1 fused GEMM + bias + relu -> h   (M=50000, K=128, N=256)
  {
    dim3 g(NN / 16, D1 / 64);
    sage_wmma_gemm<<<g, 32, 0, stream>>>(agg1, cnt, x, W1l, W1r, b1, h,
                                         D0, D1, 1);
  }

  // Layer 2 scatter: agg2 += h[src]  (degrees already in cnt)
  {
    long total = (long)NE * (D1 / 4);
    sage_scatter<<<(int)((total + 255) / 256), 256, 0, stream>>>(
        h, ei, agg2, cnt, D1, 0);
  }

  // Layer 2 fused GEMM + bias -> out   (M=50000, K=256, N=64)
  {
    dim3 g(NN / 16, D2 / 64);
    sage_wmma_gemm<<<g, 32, 0, stream>>>(agg2, cnt, h, W2l, W2r, b2, out,
                                         D1, D2, 0);
  }
}